// VectorQuantizer_39170101740093
// MI455X (gfx1250) — compile-verified
//
#include <hip/hip_runtime.h>

typedef __attribute__((ext_vector_type(2))) float v2f;
typedef __attribute__((ext_vector_type(8))) float v8f;

#define EMB        256           // C (embedding dim)
#define KCB        8192          // codebook entries
#define NTOT       32768         // B*H*W rows
#define CH_STRIDE  1024          // H*W: stride between channels in z/out
#define B_STRIDE   262144        // 256*1024: stride between images
#define WAVES      8
#define ROWS_PER_BLOCK (WAVES * 16)     // 128
#define NBLOCKS    (NTOT / ROWS_PER_BLOCK)  // 256
#define KTILES     (KCB / 16)    // 512
#define KCHUNKS    (EMB / 4)     // 64 wmma k-steps per tile
#define LPAD       260           // padded LDS row stride (floats): bank-conflict free
#define NBUF       3             // triple buffer -> single barrier per tile

// Async copy: global (per-lane 64-bit addr) -> LDS (per-lane 32-bit offset), 16B.
__device__ __forceinline__ void async_b128(const float* gsrc, float* lds_dst) {
    const unsigned ldsoff = (unsigned)(uintptr_t)lds_dst;   // LDS offset in low 32b
    asm volatile("global_load_async_to_lds_b128 %0, %1, off"
                 :: "v"(ldsoff), "v"(gsrc) : "memory");
}

// ---------------------------------------------------------------------------
// Pass 1: codebook row squared norms -> d_ws[0..8191]
// ---------------------------------------------------------------------------
__global__ __launch_bounds__(256) void vq_cbnorm(const float* __restrict__ cb,
                                                 float* __restrict__ norms) {
    __shared__ float red[256];
    const int row = blockIdx.x;
    const float v = cb[(size_t)row * EMB + threadIdx.x];
    red[threadIdx.x] = v * v;
    __syncthreads();
    for (int s = 128; s > 0; s >>= 1) {
        if (threadIdx.x < (unsigned)s) red[threadIdx.x] += red[threadIdx.x + s];
        __syncthreads();
    }
    if (threadIdx.x == 0) norms[row] = red[0];
}

// ---------------------------------------------------------------------------
// Pass 2: main VQ kernel. Per wave: 16 z rows (A in 128 VGPRs). Codebook tiles
// (16 entries x 256 f32) are triple-buffered in LDS via async global->LDS DMA,
// shared by all 8 waves (8x L2 traffic cut, one barrier per tile). fp32 WMMA
// 16x16x4 with dual interleaved accumulator chains; running argmin of
// ||e||^2 - 2 z.e; then z_q gather / out / indices / loss partial.
// ---------------------------------------------------------------------------
__global__ __launch_bounds__(256) void vq_main(const float* __restrict__ zin,
                                               const float* __restrict__ cb,
                                               const float* __restrict__ cbnorm,
                                               float* __restrict__ out,
                                               int* __restrict__ idxOut,
                                               float* __restrict__ blockLoss) {
    __shared__ float sBuf[NBUF][16][LPAD];  // padded: lane(col,half) -> unique banks
    __shared__ int   sIdx[WAVES][16];
    __shared__ float sLoss[256];

    const int lane = threadIdx.x & 31;
    const int wave = threadIdx.x >> 5;
    const int half = lane >> 4;      // 0: K=0,1 / M=0..7 rows; 1: K=2,3 / M=8..15
    const int col  = lane & 15;
    const int rowBase = (blockIdx.x * WAVES + wave) * 16;

    // async-copy work assignment: thread -> (tile row, 16B segment)
    const int cpRow = threadIdx.x >> 4;   // 0..15
    const int cpSeg = threadIdx.x & 15;   // 0..15 (x4 segments of 16B per row)

    // ---- Load A fragments (z rows) once: a[kk] = z[rowBase+col][4kk+2h .. +1]
    const int nA   = rowBase + col;
    const float* zrow = zin + (size_t)(nA >> 10) * B_STRIDE + (nA & 1023);
    v2f a[KCHUNKS];
#pragma unroll
    for (int kk = 0; kk < KCHUNKS; ++kk) {
        const int c0 = 4 * kk + 2 * half;
        a[kk].x = zrow[(size_t)c0 * CH_STRIDE];
        a[kk].y = zrow[(size_t)(c0 + 1) * CH_STRIDE];
    }

    float minVal[8];
    int   minIdx[8];
#pragma unroll
    for (int v = 0; v < 8; ++v) { minVal[v] = 3.4e38f; minIdx[v] = 0; }

    // ---- prologue: kick off tile 0 into buffer 0
    const float* gRow = cb + (size_t)cpRow * EMB;   // this thread's copy row, tile 0
#pragma unroll
    for (int j = 0; j < 4; ++j)
        async_b128(gRow + 4 * (cpSeg + 16 * j), &sBuf[0][cpRow][4 * (cpSeg + 16 * j)]);

    // ---- K loop over codebook tiles (16 entries each) ----
    int rd = 0;
    for (int kt = 0; kt < KTILES; ++kt) {
        int wr = rd + 1; if (wr == NBUF) wr = 0;
        if (kt + 1 < KTILES) {
            // issue tile kt+1 (its reader freed this buffer 2 iterations ago)
            const float* g = gRow + (size_t)(kt + 1) * 16 * EMB;
#pragma unroll
            for (int j = 0; j < 4; ++j)
                async_b128(g + 4 * (cpSeg + 16 * j),
                           &sBuf[wr][cpRow][4 * (cpSeg + 16 * j)]);
            // in-order completion: <=4 outstanding => tile kt's 4 are done
            asm volatile("s_wait_asynccnt 0x4" ::: "memory");
        } else {
            asm volatile("s_wait_asynccnt 0x0" ::: "memory");
        }
        __syncthreads();   // all waves: buffer rd fully in LDS (single barrier/tile)

        const int nIdx = kt * 16 + col;          // this lane's codebook column
        const float nrm = cbnorm[nIdx];
        v8f acc0 = {};
        v8f acc1 = {};
#pragma unroll
        for (int kk = 0; kk < KCHUNKS; kk += 2) {   // dual independent WMMA chains
            const v2f b0 = *(const v2f*)&sBuf[rd][col][4 * kk + 2 * half];
            const v2f b1 = *(const v2f*)&sBuf[rd][col][4 * (kk + 1) + 2 * half];
            acc0 = __builtin_amdgcn_wmma_f32_16x16x4_f32(
                false, a[kk], false, b0, (short)0, acc0, false, false);
            acc1 = __builtin_amdgcn_wmma_f32_16x16x4_f32(
                false, a[kk + 1], false, b1, (short)0, acc1, false, false);
        }
#pragma unroll
        for (int v = 0; v < 8; ++v) {
            const float s = fmaf(-2.0f, acc0[v] + acc1[v], nrm);  // ||e||^2 - 2 z.e
            if (s < minVal[v]) { minVal[v] = s; minIdx[v] = nIdx; }
        }
        rd = wr;
    }

    // ---- argmin reduce across the 16 lanes of each half (row m = v + 8*half)
#pragma unroll
    for (int v = 0; v < 8; ++v) {
#pragma unroll
        for (int mask = 8; mask >= 1; mask >>= 1) {
            const float ov = __shfl_xor(minVal[v], mask, 32);
            const int   oi = __shfl_xor(minIdx[v], mask, 32);
            if (ov < minVal[v] || (ov == minVal[v] && oi < minIdx[v])) {
                minVal[v] = ov; minIdx[v] = oi;
            }
        }
    }

    if (col == 0) {
#pragma unroll
        for (int v = 0; v < 8; ++v) sIdx[wave][v + 8 * half] = minIdx[v];
    }
    __syncthreads();

    // ---- indices output (int32 view of d_out tail)
    if (lane < 16) idxOut[rowBase + lane] = sIdx[wave][lane];

    // ---- epilogue: z_q gather + out store + loss partial
    // lane handles row m = col, channels c = half, half+2, ...
    const int nn = rowBase + col;
    const int bb = nn >> 10;
    const int io = nn & 1023;
    const float* zr   = zin + (size_t)bb * B_STRIDE + io;
    float*       orow = out + (size_t)bb * B_STRIDE + io;
    const float* qrow = cb + (size_t)sIdx[wave][col] * EMB;

    float lacc = 0.0f;
    for (int c = half; c < EMB; c += 2) {
        const float q  = qrow[c];
        const float zv = zr[(size_t)c * CH_STRIDE];
        const float d  = q - zv;
        lacc = fmaf(d, d, lacc);
        orow[(size_t)c * CH_STRIDE] = q;
    }

    // ---- deterministic block loss reduction
    sLoss[threadIdx.x] = lacc;
    __syncthreads();
    for (int s = 128; s > 0; s >>= 1) {
        if (threadIdx.x < (unsigned)s) sLoss[threadIdx.x] += sLoss[threadIdx.x + s];
        __syncthreads();
    }
    if (threadIdx.x == 0) blockLoss[blockIdx.x] = sLoss[0];
}

// ---------------------------------------------------------------------------
// Pass 3: deterministic final loss reduction (256 block partials -> scalar)
// loss = (1 + BETA) * mean((z_q - z)^2),  BETA = 0.25, mean over 8,388,608
// ---------------------------------------------------------------------------
__global__ __launch_bounds__(256) void vq_loss_final(const float* __restrict__ blockLoss,
                                                     float* __restrict__ lossOut) {
    __shared__ float red[256];
    red[threadIdx.x] = blockLoss[threadIdx.x];
    __syncthreads();
    for (int s = 128; s > 0; s >>= 1) {
        if (threadIdx.x < (unsigned)s) red[threadIdx.x] += red[threadIdx.x + s];
        __syncthreads();
    }
    if (threadIdx.x == 0) lossOut[0] = red[0] * (1.25f / 8388608.0f);
}

// ---------------------------------------------------------------------------
extern "C" void kernel_launch(void* const* d_in, const int* in_sizes, int n_in,
                              void* d_out, int out_size, void* d_ws, size_t ws_size,
                              hipStream_t stream) {
    (void)in_sizes; (void)n_in; (void)out_size; (void)ws_size;
    const float* z  = (const float*)d_in[0];    // (32,256,32,32) fp32
    const float* cb = (const float*)d_in[1];    // (8192,256) fp32

    float* out      = (float*)d_out;                    // 8,388,608 f32
    float* lossOut  = (float*)d_out + 8388608;          // 1 f32
    int*   idxOut   = (int*)((float*)d_out + 8388609);  // 32768 int32

    float* cbnorm    = (float*)d_ws;             // 8192 f32
    float* blockLoss = (float*)d_ws + KCB;       // 256 f32

    vq_cbnorm<<<KCB, 256, 0, stream>>>(cb, cbnorm);
    vq_main<<<NBLOCKS, 256, 0, stream>>>(z, cb, cbnorm, out, idxOut, blockLoss);
    vq_loss_final<<<1, 256, 0, stream>>>(blockLoss, lossOut);
}